// SparseInsGNBNIN_25683904430826
// MI455X (gfx1250) — compile-verified
//
#include <hip/hip_runtime.h>
#include <stddef.h>

// ---------------------------------------------------------------------------
// Segmented GroupNorm (N=2M x C=64, G=32, S~100), fp32.
// Bandwidth-bound: ~1.55 GB traffic -> ~66us floor at 23.3 TB/s. No matmul
// structure (random per-row segments) => WMMA inapplicable; the CDNA5 paths
// that matter are the 320KB-WGP LDS atomic reduction, async global<->LDS
// copies (ASYNCcnt), and wave32 16-lane-per-row b128 streaming with NT hints.
// ---------------------------------------------------------------------------

typedef float f4 __attribute__((ext_vector_type(4)));
typedef float f2 __attribute__((ext_vector_type(2)));

constexpr int GRP    = 32;     // groups
constexpr int CCH    = 64;     // channels
constexpr int S_LDS  = 128;    // LDS-accumulator capacity (segments), fast path
constexpr int S_ABS  = 4096;   // absolute fallback capacity (segments)
constexpr int SLICE_FLOATS = S_LDS * GRP * 2 + S_LDS;   // s1 | s2 | cnt = 8320
constexpr float EPSV = 1e-5f;

// ---------------------------------------------------------------------------
// Pass 1 helpers: one row = 16 lanes x float4 (wave32 covers 2 rows = 512B).
// ---------------------------------------------------------------------------
__device__ __forceinline__ void row_accum(const f4& x, int s, int c4,
                                          float* __restrict__ s1a,
                                          float* __restrict__ s2a,
                                          float* __restrict__ cnta)
{
    const float a0 = x.x + x.y;
    const float a1 = x.z + x.w;
    const float q0 = x.x * x.x + x.y * x.y;
    const float q1 = x.z * x.z + x.w * x.w;
    const int gi = s * GRP + 2 * c4;
    atomicAdd(&s1a[gi],     a0);
    atomicAdd(&s1a[gi + 1], a1);
    atomicAdd(&s2a[gi],     q0);
    atomicAdd(&s2a[gi + 1], q1);
    if (c4 == 0) atomicAdd(&cnta[s], 1.0f);
}

template <bool LDS_MODE>
__device__ __forceinline__ void reduce_loop(const f4* __restrict__ feat4,
                                            const int* __restrict__ seg,
                                            int nrows, int slot, int nslots, int c4,
                                            float* __restrict__ s1a,
                                            float* __restrict__ s2a,
                                            float* __restrict__ cnta)
{
    int r = slot;
    // 2x unrolled: issue both seg + both b128 loads before any LDS atomics,
    // doubling outstanding loads per wave (latency hiding at 23.3 TB/s).
    for (; r + nslots < nrows; r += 2 * nslots) {
        const int s0 = seg[r];
        const int s1 = seg[r + nslots];
        const f4 x0 = feat4[(size_t)r * 16 + c4];
        const f4 x1 = feat4[(size_t)(r + nslots) * 16 + c4];
        row_accum(x0, s0, c4, s1a, s2a, cnta);
        row_accum(x1, s1, c4, s1a, s2a, cnta);
    }
    if (r < nrows) {
        const int s0 = seg[r];
        const f4 x0 = feat4[(size_t)r * 16 + c4];
        row_accum(x0, s0, c4, s1a, s2a, cnta);
    }
    (void)0;
}

// ---------------------------------------------------------------------------
// Pass 1: per-(segment, group) sum / sumsq / row-count reduction.
// Fast path (S<=128): ds_add_f32 into a 33KB LDS accumulator (320KB WGP LDS
// -> ~9 blocks/WGP), then async LDS->global flush of a private slice.
// No global atomics on the fast path at all.
// ---------------------------------------------------------------------------
__global__ __launch_bounds__(256)
void k_reduce(const float* __restrict__ feat, const int* __restrict__ seg,
              const int* __restrict__ pS, int nrows,
              float* __restrict__ slices,
              float* __restrict__ s1f, float* __restrict__ s2f,
              float* __restrict__ cntf)
{
    const int S = *pS;
    __shared__ __align__(16) float acc[SLICE_FLOATS];   // [s1:4096 | s2:4096 | cnt:128]

    const int  tid      = threadIdx.x;
    const int  c4       = tid & 15;
    const int  slot     = (blockIdx.x * 256 + tid) >> 4;
    const int  nslots   = (gridDim.x * 256) >> 4;
    const f4*  feat4    = (const f4*)feat;

    if (S <= S_LDS) {
        for (int i = tid; i < SLICE_FLOATS; i += 256) acc[i] = 0.0f;
        __syncthreads();

        reduce_loop<true>(feat4, seg, nrows, slot, nslots, c4,
                          acc, acc + S_LDS * GRP, acc + 2 * S_LDS * GRP);

        __syncthreads();
        // Async LDS -> global flush (SLICE_FLOATS/4 = 2080 x 16B chunks).
        float* slice = slices + (size_t)blockIdx.x * SLICE_FLOATS;
        const unsigned lds_base = (unsigned)(size_t)&acc[0];
        for (int i = tid; i < SLICE_FLOATS / 4; i += 256) {
            const unsigned goff  = (unsigned)i * 16u;
            const unsigned laddr = lds_base + (unsigned)i * 16u;
            asm volatile("global_store_async_from_lds_b128 %0, %1, %2"
                         :: "v"(goff), "v"(laddr), "s"(slice) : "memory");
        }
        asm volatile("s_wait_asynccnt 0" ::: "memory");
    } else {
        // Correctness fallback (never hit for the harness S=100): global atomics.
        reduce_loop<false>(feat4, seg, nrows, slot, nslots, c4, s1f, s2f, cntf);
    }
}

// ---------------------------------------------------------------------------
// Pass 2: combine per-block slices -> (mean, inv_std) per (segment, group).
// 3200 live entries; coalesced across the (s,g) thread index per slice.
// ---------------------------------------------------------------------------
__global__ __launch_bounds__(256)
void k_stats(const int* __restrict__ pS, int nblk,
             const float* __restrict__ slices,
             const float* __restrict__ s1f, const float* __restrict__ s2f,
             const float* __restrict__ cntf,
             f2* __restrict__ mean_inv)
{
    const int S = *pS;
    const int i = blockIdx.x * 256 + threadIdx.x;        // flat (s,g)
    if (i >= S * GRP) return;
    const int s = i >> 5;

    float s1 = 0.0f, s2 = 0.0f, rows = 0.0f;
    if (S <= S_LDS) {
        for (int b = 0; b < nblk; ++b) {
            const float* sl = slices + (size_t)b * SLICE_FLOATS;
            s1   += sl[i];
            s2   += sl[S_LDS * GRP + i];
            rows += sl[2 * S_LDS * GRP + s];
        }
    } else {
        s1 = s1f[i]; s2 = s2f[i]; rows = cntf[s];
    }
    const float denom = fmaxf(rows * 2.0f, 1.0f);        // cnt = rows * Cg, guarded
    const float mean  = s1 / denom;
    const float var   = s2 / denom - mean * mean;
    const float inv   = rsqrtf(var + EPSV);
    f2 mi; mi.x = mean; mi.y = inv;
    mean_inv[i] = mi;
}

// ---------------------------------------------------------------------------
// Pass 3 helper: normalize loop, compile-time table selection so the LDS
// path keeps addrspace(3) provenance (ds_load, not flat).
// ---------------------------------------------------------------------------
template <bool LDS_MODE>
__device__ __forceinline__ void apply_loop(const f4* __restrict__ feat4,
                                           const int* __restrict__ seg,
                                           int nrows, int slot, int nslots, int c4,
                                           const f2* st, const f2* __restrict__ mi,
                                           const f4& gm, const f4& bt,
                                           f4* __restrict__ out4)
{
    auto norm1 = [&](int r) {
        const int s = seg[r];
        const f4 x = __builtin_nontemporal_load(&feat4[(size_t)r * 16 + c4]);
        f2 m0, m1;
        if constexpr (LDS_MODE) {
            m0 = st[s * GRP + 2 * c4];
            m1 = st[s * GRP + 2 * c4 + 1];
        } else {
            m0 = mi[s * GRP + 2 * c4];
            m1 = mi[s * GRP + 2 * c4 + 1];
        }
        f4 y;
        y.x = (x.x - m0.x) * m0.y;
        y.y = (x.y - m0.x) * m0.y;
        y.z = (x.z - m1.x) * m1.y;
        y.w = (x.w - m1.x) * m1.y;
        y = y * gm + bt;
        __builtin_nontemporal_store(y, &out4[(size_t)r * 16 + c4]);
    };

    int r = slot;
    for (; r + nslots < nrows; r += 2 * nslots) {
        // 2x unroll with loads hoisted ahead of compute for MLP.
        const int ra = r, rb = r + nslots;
        const int sa = seg[ra], sb = seg[rb];
        const f4 xa = __builtin_nontemporal_load(&feat4[(size_t)ra * 16 + c4]);
        const f4 xb = __builtin_nontemporal_load(&feat4[(size_t)rb * 16 + c4]);
        f2 a0, a1, b0, b1;
        if constexpr (LDS_MODE) {
            a0 = st[sa * GRP + 2 * c4]; a1 = st[sa * GRP + 2 * c4 + 1];
            b0 = st[sb * GRP + 2 * c4]; b1 = st[sb * GRP + 2 * c4 + 1];
        } else {
            a0 = mi[sa * GRP + 2 * c4]; a1 = mi[sa * GRP + 2 * c4 + 1];
            b0 = mi[sb * GRP + 2 * c4]; b1 = mi[sb * GRP + 2 * c4 + 1];
        }
        f4 ya, yb;
        ya.x = (xa.x - a0.x) * a0.y;  ya.y = (xa.y - a0.x) * a0.y;
        ya.z = (xa.z - a1.x) * a1.y;  ya.w = (xa.w - a1.x) * a1.y;
        yb.x = (xb.x - b0.x) * b0.y;  yb.y = (xb.y - b0.x) * b0.y;
        yb.z = (xb.z - b1.x) * b1.y;  yb.w = (xb.w - b1.x) * b1.y;
        ya = ya * gm + bt;
        yb = yb * gm + bt;
        __builtin_nontemporal_store(ya, &out4[(size_t)ra * 16 + c4]);
        __builtin_nontemporal_store(yb, &out4[(size_t)rb * 16 + c4]);
    }
    if (r < nrows) norm1(r);
}

// ---------------------------------------------------------------------------
// Pass 3: normalize. Stage the 25.6KB (mean,inv) table into LDS with async
// global->LDS copies, then stream features NT-in / NT-out (512MB streams
// >> 192MB L2; keep seg + stats resident instead).
// ---------------------------------------------------------------------------
__global__ __launch_bounds__(256)
void k_apply(const float* __restrict__ feat, const int* __restrict__ seg,
             const float* __restrict__ gamma, const float* __restrict__ beta,
             const int* __restrict__ pS, int nrows,
             const f2* __restrict__ mean_inv, float* __restrict__ out)
{
    const int S = *pS;
    __shared__ __align__(16) f2 st[S_LDS * GRP];         // 32KB stats table

    const int tid    = threadIdx.x;
    const int c4     = tid & 15;
    const int slot   = (blockIdx.x * 256 + tid) >> 4;
    const int nslots = (gridDim.x * 256) >> 4;
    const f4* feat4  = (const f4*)feat;
    f4*       out4   = (f4*)out;
    const f4  gm = ((const f4*)gamma)[c4];
    const f4  bt = ((const f4*)beta)[c4];

    if (S <= S_LDS) {
        const unsigned lds_base = (unsigned)(size_t)&st[0];
        const int chunks = S * 16;                        // (S*GRP*8)/16 -> 16B chunks
        for (int i = tid; i < chunks; i += 256) {
            const unsigned goff  = (unsigned)i * 16u;
            const unsigned laddr = lds_base + (unsigned)i * 16u;
            asm volatile("global_load_async_to_lds_b128 %0, %1, %2"
                         :: "v"(laddr), "v"(goff), "s"(mean_inv) : "memory");
        }
        asm volatile("s_wait_asynccnt 0" ::: "memory");
        __syncthreads();
        apply_loop<true>(feat4, seg, nrows, slot, nslots, c4, st, mean_inv, gm, bt, out4);
    } else {
        apply_loop<false>(feat4, seg, nrows, slot, nslots, c4, st, mean_inv, gm, bt, out4);
    }
}

// ---------------------------------------------------------------------------
extern "C" void kernel_launch(void* const* d_in, const int* in_sizes, int n_in,
                              void* d_out, int out_size, void* d_ws, size_t ws_size,
                              hipStream_t stream)
{
    const float* feat  = (const float*)d_in[0];
    const float* gamma = (const float*)d_in[1];
    const float* beta  = (const float*)d_in[2];
    const int*   seg   = (const int*)d_in[3];
    const int*   pS    = (const int*)d_in[4];
    const int    nrows = in_sizes[0] / CCH;

    // Workspace layout (all offsets 16B aligned):
    //   [ s1f: S_ABS*G ][ s2f: S_ABS*G ][ cntf: S_ABS ][ mean_inv: S_ABS*G f2 ][ slices ... ]
    char* ws = (char*)d_ws;
    size_t off = 0;
    float* s1f  = (float*)(ws + off); off += (size_t)S_ABS * GRP * sizeof(float);
    float* s2f  = (float*)(ws + off); off += (size_t)S_ABS * GRP * sizeof(float);
    float* cntf = (float*)(ws + off); off += (size_t)S_ABS * sizeof(float);
    const size_t fallback_bytes = off;                    // zeroed every launch
    f2* mean_inv = (f2*)(ws + off);   off += (size_t)S_ABS * GRP * sizeof(f2);
    float* slices = (float*)(ws + off);

    const size_t slice_bytes = (size_t)SLICE_FLOATS * sizeof(float);
    size_t avail = (ws_size > off) ? (ws_size - off) : 0;
    int nblk1 = (int)(avail / slice_bytes);
    if (nblk1 > 512) nblk1 = 512;
    if (nblk1 < 1)  nblk1 = 1;

    // Zero fallback accumulators (used only when S > 128).
    hipMemsetAsync(d_ws, 0, fallback_bytes, stream);

    k_reduce<<<nblk1, 256, 0, stream>>>(feat, seg, pS, nrows, slices, s1f, s2f, cntf);
    k_stats<<<(S_ABS * GRP) / 256, 256, 0, stream>>>(pS, nblk1, slices, s1f, s2f, cntf, mean_inv);
    k_apply<<<2048, 256, 0, stream>>>(feat, seg, gamma, beta, pS, nrows, mean_inv, (float*)d_out);
}